// FFTModel_74586402062715
// MI455X (gfx1250) — compile-verified
//
#include <hip/hip_runtime.h>

// Stockham FFT, radix-16 fused passes on gfx1250 (wave32).
// 23 stages = 6 passes: s = 0,4,8,12,16 (k=4) and s = 20 (k=3).
// Memory-bound: 6 x 134MB ping-pong traffic, L2-resident (192MB).

#define LOGN 23
#define N_TOTAL (1u << LOGN)

typedef int v2i __attribute__((vector_size(8)));  // 64-bit payload type for async LDS ops
typedef __attribute__((address_space(1))) v2i* gptr_v2i;
typedef __attribute__((address_space(3))) v2i* lptr_v2i;

__device__ __forceinline__ constexpr int brev(int t, int K) {
  int r = 0;
  for (int i = 0; i < K; ++i) r |= ((t >> i) & 1) << (K - 1 - i);
  return r;
}

template <int S, int K, bool FIRST, bool STAGED>
__global__ __launch_bounds__(256)
void fft_pass(const void* __restrict__ src, float2* __restrict__ dst,
              const float2* __restrict__ weight) {
  constexpr int R = 1 << K;               // radix (complex points per thread)
  constexpr int BMASK_S = (1 << S) - 1;
  const int b = blockIdx.x * 256 + threadIdx.x;   // base index in [0, 2^(LOGN-K))
  const int blow = b & BMASK_S;

  // ---- load: 2^K chunks strided by N/2^K, each fully coalesced ----
  float2 y[R];
  if (FIRST) {
    const float* in = (const float*)src;   // real input, imag = 0
#pragma unroll
    for (int t = 0; t < R; ++t) {
      float v = in[((size_t)t << (LOGN - K)) + (size_t)b];
      y[t] = make_float2(v, 0.0f);
    }
  } else {
    const float2* in = (const float2*)src;
#pragma unroll
    for (int t = 0; t < R; ++t)
      y[t] = in[((size_t)t << (LOGN - K)) + (size_t)b];
  }

  // ---- K butterfly sub-stages, in registers (DIF ordering over t) ----
#pragma unroll
  for (int r = 0; r < K; ++r) {
    const int g = S + r;                   // global stage index
    const int wshift = (LOGN - 1) - g;     // twiddle stride = 2^wshift
    const int half = 1 << (K - 1 - r);     // pair distance in t
#pragma unroll
    for (int t0 = 0; t0 < R; ++t0) {
      if (t0 & half) continue;             // lower element of each pair
      // E = bit-reverse of the r already-consumed top bits of t0
      int E = 0;
#pragma unroll
      for (int rp = 0; rp < r; ++rp)
        E |= ((t0 >> (K - 1 - rp)) & 1) << rp;
      const int j = (E << S) | blow;       // column index of global stage g
      const float2 w = weight[(size_t)j << wshift];
      const float2 a = y[t0];
      const float2 c = y[t0 + half];
      const float pr = c.x * w.x - c.y * w.y;
      const float pi = c.x * w.y + c.y * w.x;
      y[t0]        = make_float2(a.x + pr, a.y + pi);
      y[t0 + half] = make_float2(a.x - pr, a.y - pi);
    }
  }

  // ---- store: q = (b>>S)<<(S+K) | rev_K(t)<<S | (b & (2^S-1)) ----
  if (!STAGED) {
    // S >= 5: lanes (consecutive b) give consecutive q per t -> coalesced
    const size_t obase = (((size_t)(b >> S)) << (S + K)) | (size_t)blow;
#pragma unroll
    for (int t = 0; t < R; ++t)
      dst[obase + ((size_t)brev(t, K) << S)] = y[t];
  } else {
    // S < 5: scatter within the block tile; transpose via padded LDS, then
    // drain with coalesced async stores from LDS (CDNA5 ASYNCcnt path).
    constexpr int TILE = 256 * R;                 // complex elements per block
    __shared__ float2 lbuf[TILE + (TILE >> 4)];   // +1 complex pad per 16
    const int bl = threadIdx.x;
#pragma unroll
    for (int t = 0; t < R; ++t) {
      const int lidx = ((bl >> S) << (S + K)) | (brev(t, K) << S) | (bl & BMASK_S);
      lbuf[lidx + (lidx >> 4)] = y[t];
    }
    __syncthreads();
    const size_t out_base = (size_t)blockIdx.x * TILE;  // contiguous block tile
#pragma unroll
    for (int c = 0; c < R; ++c) {
      const int idx = c * 256 + threadIdx.x;
      float2* gp = dst + out_base + (size_t)idx;
      float2* lp = &lbuf[idx + (idx >> 4)];
#if defined(__gfx1250__) && __has_builtin(__builtin_amdgcn_global_store_async_from_lds_b64)
      __builtin_amdgcn_global_store_async_from_lds_b64(
          (gptr_v2i)gp, (lptr_v2i)lp, /*offset=*/0, /*cpol=*/0);
#else
      *gp = *lp;
#endif
    }
    // s_endpgm performs an implicit wait-idle (all counters, incl. ASYNCcnt)
  }
}

extern "C" void kernel_launch(void* const* d_in, const int* in_sizes, int n_in,
                              void* d_out, int out_size, void* d_ws, size_t ws_size,
                              hipStream_t stream) {
  const float*  input  = (const float*)d_in[0];          // (N,) f32
  const float2* weight = (const float2*)d_in[1];         // (N/2, 2) f32
  float2* out = (float2*)d_out;                          // (N, 2) f32 out
  float2* ws  = (float2*)d_ws;                           // needs 64 MiB

  const dim3 blk(256);
  const int blocks_k4 = (N_TOTAL >> 4) / 256;  // 2048
  const int blocks_k3 = (N_TOTAL >> 3) / 256;  // 4096

  // stages 0-3   : input -> ws   (LDS-staged stores, async drain)
  fft_pass<0, 4, true,  true ><<<blocks_k4, blk, 0, stream>>>(input, ws, weight);
  // stages 4-7   : ws -> out     (LDS-staged stores, async drain)
  fft_pass<4, 4, false, true ><<<blocks_k4, blk, 0, stream>>>(ws, out, weight);
  // stages 8-11  : out -> ws     (direct coalesced stores)
  fft_pass<8, 4, false, false><<<blocks_k4, blk, 0, stream>>>(out, ws, weight);
  // stages 12-15 : ws -> out
  fft_pass<12, 4, false, false><<<blocks_k4, blk, 0, stream>>>(ws, out, weight);
  // stages 16-19 : out -> ws
  fft_pass<16, 4, false, false><<<blocks_k4, blk, 0, stream>>>(out, ws, weight);
  // stages 20-22 : ws -> out     (radix-8 final pass)
  fft_pass<20, 3, false, false><<<blocks_k3, blk, 0, stream>>>(ws, out, weight);
}